// SSMBlock_59261958750762
// MI455X (gfx1250) — compile-verified
//
#include <hip/hip_runtime.h>
#include <hip/hip_bf16.h>
#include <math.h>

typedef __attribute__((ext_vector_type(2))) float v2f;
typedef __attribute__((ext_vector_type(8))) float v8f;

#define C_CH   16
#define P_LEN  256
#define DM_    512
#define DI_    1024
#define DSTATE 16
#define DDT    32
#define MROWS  (C_CH * P_LEN)   // 4096

// ---------------------------------------------------------------- helpers ---

__device__ __forceinline__ float gelu_tanh(float x) {
    // jax.nn.gelu default (approximate=True)
    const float k0 = 0.7978845608028654f;   // sqrt(2/pi)
    float x3 = x * x * x;
    return 0.5f * x * (1.0f + tanhf(k0 * (x + 0.044715f * x3)));
}

__device__ __forceinline__ float softplus_f(float x) {
    return (x > 20.0f) ? x : log1pf(expf(x));
}

__device__ __forceinline__ v8f wmma4(v2f a, v2f b, v8f c) {
    // V_WMMA_F32_16X16X4_F32  (exact fp32 path)
    // 8 args: (neg_a, A, neg_b, B, c_mod, C, reuse_a, reuse_b)
    return __builtin_amdgcn_wmma_f32_16x16x4_f32(
        false, a, false, b, (short)0, c, false, false);
}

// -------------------------------------------------- generic WMMA f32 GEMM ---
// out[M,N] = act( A[M,(lda)] (cols 0..K-1) @ W[N,K]^T + bias[N] )
// block = 256 threads (8 wave32 waves); block tile 256(M) x 64(N);
// wave tile 32(M) x 64(N): 2 A fragments x 4 B fragments -> 8 WMMAs per
// 6 loads (each B fragment reused twice, halving weight traffic).
// EPI: 0 = none, 1 = gelu, 2 = softplus

template <int EPI>
__global__ __launch_bounds__(256)
void gemm_wmma_kernel(const float* __restrict__ A, int lda,
                      const float* __restrict__ W,
                      const float* __restrict__ bias,
                      float* __restrict__ out,
                      int N, int K)
{
    const int lane = threadIdx.x & 31;
    const int wave = threadIdx.x >> 5;
    const int lid  = lane & 15;       // row/col within 16
    const int hi   = lane >> 4;       // half-wave select

    const int n0 = blockIdx.x * 64;
    const int r0 = blockIdx.y * 256 + wave * 32;

    const float* arow0 = A + (size_t)(r0 +      lid) * lda;
    const float* arow1 = A + (size_t)(r0 + 16 + lid) * lda;
    const float* wrow0 = W + (size_t)(n0 +  0 + lid) * K;
    const float* wrow1 = W + (size_t)(n0 + 16 + lid) * K;
    const float* wrow2 = W + (size_t)(n0 + 32 + lid) * K;
    const float* wrow3 = W + (size_t)(n0 + 48 + lid) * K;

    v8f acc00 = {}, acc01 = {}, acc02 = {}, acc03 = {};
    v8f acc10 = {}, acc11 = {}, acc12 = {}, acc13 = {};

    #pragma unroll 2
    for (int k = 0; k < K; k += 4) {
        // A frag: lanes 0-15 hold K={k,k+1}, lanes 16-31 hold K={k+2,k+3}
        v2f a0 = *(const v2f*)(arow0 + k + 2 * hi);
        v2f a1 = *(const v2f*)(arow1 + k + 2 * hi);
        v2f b0 = *(const v2f*)(wrow0 + k + 2 * hi);
        v2f b1 = *(const v2f*)(wrow1 + k + 2 * hi);
        v2f b2 = *(const v2f*)(wrow2 + k + 2 * hi);
        v2f b3 = *(const v2f*)(wrow3 + k + 2 * hi);
        acc00 = wmma4(a0, b0, acc00);
        acc10 = wmma4(a1, b0, acc10);
        acc01 = wmma4(a0, b1, acc01);
        acc11 = wmma4(a1, b1, acc11);
        acc02 = wmma4(a0, b2, acc02);
        acc12 = wmma4(a1, b2, acc12);
        acc03 = wmma4(a0, b3, acc03);
        acc13 = wmma4(a1, b3, acc13);
    }

    #pragma unroll
    for (int m = 0; m < 2; ++m) {
        const int rbase = r0 + m * 16;
        #pragma unroll
        for (int j = 0; j < 4; ++j) {
            v8f acc;
            if (m == 0) acc = (j == 0) ? acc00 : (j == 1) ? acc01 : (j == 2) ? acc02 : acc03;
            else        acc = (j == 0) ? acc10 : (j == 1) ? acc11 : (j == 2) ? acc12 : acc13;
            const int col = n0 + j * 16 + lid;
            const float bb = bias ? bias[col] : 0.0f;
            #pragma unroll
            for (int g = 0; g < 8; ++g) {
                // C/D layout: VGPR g -> M = g (lanes 0-15) / g+8 (lanes 16-31)
                const int row = rbase + g + 8 * hi;
                float v = acc[g] + bb;
                if (EPI == 1)      v = gelu_tanh(v);
                else if (EPI == 2) v = softplus_f(v);
                out[(size_t)row * N + col] = v;
            }
        }
    }
}

// ----------------------------------------- conv1d as 4 accumulated GEMMs ---
// xi_conv[c,p,o] = gelu( sum_t sum_i  xi[c, clamp(p+t-2), i] * w[o,i,t] + b[o] )
// xi lives in xz (row stride 2*DI, first DI columns).  conv_wt is [t][n][k].
// One block covers exactly one chunk (256 rows), so clamping never straddles.

__global__ __launch_bounds__(256)
void conv_gemm_kernel(const float* __restrict__ xz,
                      const float* __restrict__ conv_wt,
                      const float* __restrict__ conv_b,
                      float* __restrict__ out)
{
    const int lane = threadIdx.x & 31;
    const int wave = threadIdx.x >> 5;
    const int lid  = lane & 15;
    const int hi   = lane >> 4;

    const int n0 = blockIdx.x * 64;
    const int c  = blockIdx.y;                 // one chunk per block row
    const int p0 = wave * 32 + lid;            // low 16-row group
    const int p1 = p0 + 16;                    // high 16-row group

    v8f acc00 = {}, acc01 = {}, acc02 = {}, acc03 = {};
    v8f acc10 = {}, acc11 = {}, acc12 = {}, acc13 = {};

    for (int t = 0; t < 4; ++t) {
        int pr0 = p0 + t - 2;
        int pr1 = p1 + t - 2;
        pr0 = pr0 < 0 ? 0 : (pr0 > P_LEN - 1 ? P_LEN - 1 : pr0);   // edge pad
        pr1 = pr1 < 0 ? 0 : (pr1 > P_LEN - 1 ? P_LEN - 1 : pr1);
        const float* arow0 = xz + (size_t)(c * P_LEN + pr0) * (2 * DI_);
        const float* arow1 = xz + (size_t)(c * P_LEN + pr1) * (2 * DI_);
        const float* wb    = conv_wt + (size_t)t * DI_ * DI_;
        const float* wrow0 = wb + (size_t)(n0 +  0 + lid) * DI_;
        const float* wrow1 = wb + (size_t)(n0 + 16 + lid) * DI_;
        const float* wrow2 = wb + (size_t)(n0 + 32 + lid) * DI_;
        const float* wrow3 = wb + (size_t)(n0 + 48 + lid) * DI_;

        #pragma unroll 2
        for (int k = 0; k < DI_; k += 4) {
            v2f a0 = *(const v2f*)(arow0 + k + 2 * hi);
            v2f a1 = *(const v2f*)(arow1 + k + 2 * hi);
            v2f b0 = *(const v2f*)(wrow0 + k + 2 * hi);
            v2f b1 = *(const v2f*)(wrow1 + k + 2 * hi);
            v2f b2 = *(const v2f*)(wrow2 + k + 2 * hi);
            v2f b3 = *(const v2f*)(wrow3 + k + 2 * hi);
            acc00 = wmma4(a0, b0, acc00);
            acc10 = wmma4(a1, b0, acc10);
            acc01 = wmma4(a0, b1, acc01);
            acc11 = wmma4(a1, b1, acc11);
            acc02 = wmma4(a0, b2, acc02);
            acc12 = wmma4(a1, b2, acc12);
            acc03 = wmma4(a0, b3, acc03);
            acc13 = wmma4(a1, b3, acc13);
        }
    }

    #pragma unroll
    for (int m = 0; m < 2; ++m) {
        const int pbase = wave * 32 + m * 16;
        #pragma unroll
        for (int j = 0; j < 4; ++j) {
            v8f acc;
            if (m == 0) acc = (j == 0) ? acc00 : (j == 1) ? acc01 : (j == 2) ? acc02 : acc03;
            else        acc = (j == 0) ? acc10 : (j == 1) ? acc11 : (j == 2) ? acc12 : acc13;
            const int col = n0 + j * 16 + lid;
            const float bb = conv_b[col];
            #pragma unroll
            for (int g = 0; g < 8; ++g) {
                const int row = c * P_LEN + pbase + g + 8 * hi;
                out[(size_t)row * DI_ + col] = gelu_tanh(acc[g] + bb);
            }
        }
    }
}

// ----------------------------------------------- conv weight repack kernel ---
// conv_w (DI,DI,4) -> conv_wt [t][n][k] so GEMM B-fragments are contiguous.

__global__ void repack_convw(const float* __restrict__ w, float* __restrict__ wt)
{
    int i = blockIdx.x * blockDim.x + threadIdx.x;
    if (i >= 4 * DI_ * DI_) return;
    int t   = i / (DI_ * DI_);
    int rem = i % (DI_ * DI_);
    int n = rem / DI_, k = rem % DI_;
    wt[i] = w[((size_t)n * DI_ + k) * 4 + t];
}

// ------------------------------------------------------------- scan kernel ---
// One thread per (c, d); 16-deep state in registers; B/C staged via LDS.

__global__ __launch_bounds__(256)
void scan_kernel(const float* __restrict__ xz,     // z at column offset DI
                 const float* __restrict__ xi,     // conv+gelu, [MROWS][DI]
                 const float* __restrict__ xdb,    // [MROWS][64] (dt|B|C)
                 const float* __restrict__ dtv,    // softplus'd dt, [MROWS][DI]
                 const float* __restrict__ ssm_in, // [C][DI][16]
                 const float* __restrict__ A_log,  // [DI][16]
                 const float* __restrict__ Dvec,   // [DI]
                 float* __restrict__ ygated,       // [MROWS][DI]
                 float* __restrict__ ssm_out)      // [C][DI][16]
{
    __shared__ float sB[DSTATE];
    __shared__ float sC[DSTATE];

    const int tid = threadIdx.x;
    const int gid = blockIdx.x * 256 + tid;
    const int c = gid >> 10;          // / DI
    const int d = gid & (DI_ - 1);    // % DI

    float Arow[DSTATE], s[DSTATE];
    #pragma unroll
    for (int n = 0; n < DSTATE; ++n) {
        Arow[n] = -expf(A_log[d * DSTATE + n]);           // A = -exp(A_log)
        s[n]    = ssm_in[((size_t)c * DI_ + d) * DSTATE + n];
    }
    const float Dd = Dvec[d];

    for (int p = 0; p < P_LEN; ++p) {
        const size_t r = (size_t)c * P_LEN + p;
        if (tid < 32) {
            float v = xdb[r * 64 + 32 + tid];
            if (tid < 16) sB[tid] = v; else sC[tid - 16] = v;
        }
        __syncthreads();

        const float dt = dtv[r * DI_ + d];
        const float xt = xi [r * DI_ + d];
        const float zr = xz [r * (2 * DI_) + DI_ + d];

        float y = 0.0f;
        #pragma unroll
        for (int n = 0; n < DSTATE; ++n) {
            float dA = __expf(dt * Arow[n]);              // exp(dt*A)
            s[n] = s[n] * dA + xt * dt * sB[n];           // s*dA + x*dB
            y += s[n] * sC[n];
        }
        y += Dd * xt;
        const float g = gelu_tanh(gelu_tanh(zr));         // double gelu (faithful)
        ygated[r * DI_ + d] = y * g;
        __syncthreads();
    }

    #pragma unroll
    for (int n = 0; n < DSTATE; ++n)
        ssm_out[((size_t)c * DI_ + d) * DSTATE + n] = s[n];
}

// ------------------------------------------------------------------ launch ---

extern "C" void kernel_launch(void* const* d_in, const int* in_sizes, int n_in,
                              void* d_out, int out_size, void* d_ws, size_t ws_size,
                              hipStream_t stream)
{
    const float* x       = (const float*)d_in[0];
    const float* ssm0    = (const float*)d_in[1];
    const float* in_w    = (const float*)d_in[2];
    const float* in_b    = (const float*)d_in[3];
    const float* conv_w  = (const float*)d_in[4];
    const float* conv_b  = (const float*)d_in[5];
    const float* param_w = (const float*)d_in[6];
    const float* param_b = (const float*)d_in[7];
    const float* dt_w    = (const float*)d_in[8];
    const float* dt_b    = (const float*)d_in[9];
    const float* out_w   = (const float*)d_in[10];
    const float* out_b   = (const float*)d_in[11];
    const float* A_log   = (const float*)d_in[12];
    const float* Dvec    = (const float*)d_in[13];

    float* ws      = (float*)d_ws;
    float* xz      = ws;                                    // 4096*2048
    float* conv_wt = xz      + (size_t)MROWS * 2 * DI_;     // 4*1024*1024
    float* xi      = conv_wt + (size_t)4 * DI_ * DI_;       // 4096*1024
    float* xdb     = xi      + (size_t)MROWS * DI_;         // 4096*64
    float* dtv     = xdb     + (size_t)MROWS * 64;          // 4096*1024
    float* yg      = dtv     + (size_t)MROWS * DI_;         // 4096*1024

    float* yout    = (float*)d_out;                         // [C,P,DM]
    float* ssm_out = yout + (size_t)MROWS * DM_;            // [C,DI,16]

    // 0) repack conv weights for contiguous B fragments
    repack_convw<<<(4 * DI_ * DI_ + 255) / 256, 256, 0, stream>>>(conv_w, conv_wt);

    // 1) xz = x @ in_w^T + in_b                (M=4096, N=2048, K=512)
    gemm_wmma_kernel<0><<<dim3(2 * DI_ / 64, MROWS / 256), 256, 0, stream>>>(
        x, DM_, in_w, in_b, xz, 2 * DI_, DM_);

    // 2) conv1d (4 gathered GEMMs) + bias + gelu   (N=1024, K=1024 x 4 taps)
    conv_gemm_kernel<<<dim3(DI_ / 64, C_CH), 256, 0, stream>>>(
        xz, conv_wt, conv_b, xi);

    // 3) x_db = xi @ param_w^T + param_b       (N=64, K=1024)
    gemm_wmma_kernel<0><<<dim3(64 / 64, MROWS / 256), 256, 0, stream>>>(
        xi, DI_, param_w, param_b, xdb, 64, DI_);

    // 4) dt = softplus(x_db[:, :32] @ dt_w^T + dt_b)  (N=1024, K=32, lda=64)
    gemm_wmma_kernel<2><<<dim3(DI_ / 64, MROWS / 256), 256, 0, stream>>>(
        xdb, 64, dt_w, dt_b, dtv, DI_, DDT);

    // 5) selective scan (sequential over P, parallel over C*DI)
    scan_kernel<<<(C_CH * DI_) / 256, 256, 0, stream>>>(
        xz, xi, xdb, dtv, ssm0, A_log, Dvec, yg, ssm_out);

    // 6) out = yg @ out_w^T + out_b            (N=512, K=1024)
    gemm_wmma_kernel<0><<<dim3(DM_ / 64, MROWS / 256), 256, 0, stream>>>(
        yg, DI_, out_w, out_b, yout, DM_, DI_);
}